// UpFirDnDownsample_77129022702209
// MI455X (gfx1250) — compile-verified
//
#include <hip/hip_runtime.h>

#ifndef __has_builtin
#define __has_builtin(x) 0
#endif

#define AS1 __attribute__((address_space(1)))
#define AS3 __attribute__((address_space(3)))

typedef int v4i_t __attribute__((ext_vector_type(4)));

#if __has_builtin(__builtin_amdgcn_global_load_async_to_lds_b128)
#define HAVE_ASYNC_COPY 1
#else
#define HAVE_ASYNC_COPY 0
#endif

// 16-byte global -> LDS async DMA copy (CDNA5 ASYNCcnt path).
__device__ __forceinline__ void async_copy_b128(const float* g, float* l) {
#if HAVE_ASYNC_COPY
    AS1 v4i_t* gv = (AS1 v4i_t*)(void*)g; // strip const + addrspacecast to global
    AS3 v4i_t* lv = (AS3 v4i_t*)l;        // addrspacecast to LDS
    __builtin_amdgcn_global_load_async_to_lds_b128(gv, lv, /*imm offset*/0, /*cpol*/0);
#else
    // Fallback: plain vectorized copy through VGPRs (still correct).
    *reinterpret_cast<float4*>(l) = *reinterpret_cast<const float4*>(g);
#endif
}

__device__ __forceinline__ void wait_async_copies() {
#if HAVE_ASYNC_COPY
#if __has_builtin(__builtin_amdgcn_s_wait_asynccnt)
    __builtin_amdgcn_s_wait_asynccnt(0);
#else
    asm volatile("s_wait_asynccnt 0" ::: "memory");
#endif
#endif
}

#define IN_H 64
#define IN_W 64
#define OUT_H 32
#define OUT_W 32
#define PLANE_IN (IN_H * IN_W)    // 4096 floats = 16 KB
#define PLANE_OUT (OUT_H * OUT_W) // 1024 floats

// One block per (n,c) plane. 256 threads (8 wave32s).
__global__ __launch_bounds__(256)
void upfirdn2d_down2_kernel(const float* __restrict__ x,
                            const float* __restrict__ kern,
                            float* __restrict__ out) {
    __shared__ float tile[PLANE_IN]; // 16 KB input plane
    __shared__ float wks[16];        // flipped 4x4 kernel taps

    const int tid = threadIdx.x;
    const size_t plane = blockIdx.x;
    const float* gp = x + plane * (size_t)PLANE_IN;

    // Stage the whole plane into LDS: 1024 float4 chunks, 4 per thread,
    // contiguous across lanes -> perfectly coalesced 128B async DMA.
#pragma unroll
    for (int c = 0; c < 4; ++c) {
        const int idx = c * 256 + tid; // float4 chunk index
        async_copy_b128(gp + idx * 4, &tile[idx * 4]);
    }

    // Flip the 4x4 kernel (flip both dims == reverse flat order).
    if (tid < 16) wks[tid] = kern[15 - tid];

    wait_async_copies();   // this wave's async LDS writes done
    __syncthreads();       // all waves' writes + wks visible

    // Each thread produces out[oh, 4*owq .. 4*owq+3] (one float4).
    const int owq = tid & 7;   // 0..7  -> output col quad
    const int oh  = tid >> 3;  // 0..31 -> output row
    const int wbase = 8 * owq; // leftmost input col of this strip is wbase-1

    float acc[4] = {0.f, 0.f, 0.f, 0.f};

#pragma unroll
    for (int i = 0; i < 4; ++i) {
        const int ih = 2 * oh + i - 1;                 // pad0 = 1
        const bool rok = (ih >= 0) & (ih < IN_H);
        const int ihc = ih < 0 ? 0 : (ih >= IN_H ? IN_H - 1 : ih);
        const int rbase = ihc * IN_W;

        // 10 input columns cover all 4 outputs' 4-tap windows.
        float xv[10];
#pragma unroll
        for (int c = 0; c < 10; ++c) {
            const int iw = wbase + c - 1;
            const bool ok = rok & (iw >= 0) & (iw < IN_W);
            const int iwc = iw < 0 ? 0 : (iw >= IN_W ? IN_W - 1 : iw);
            const float v = tile[rbase + iwc]; // always-valid LDS address
            xv[c] = ok ? v : 0.0f;             // zero padding at borders
        }

#pragma unroll
        for (int m = 0; m < 4; ++m) {
#pragma unroll
            for (int j = 0; j < 4; ++j) {
                acc[m] = fmaf(wks[i * 4 + j], xv[2 * m + j], acc[m]);
            }
        }
    }

    const size_t obase = plane * (size_t)PLANE_OUT + (size_t)oh * OUT_W + wbase / 2;
    float4 r;
    r.x = acc[0]; r.y = acc[1]; r.z = acc[2]; r.w = acc[3];
    *reinterpret_cast<float4*>(out + obase) = r; // coalesced 128-bit store
}

extern "C" void kernel_launch(void* const* d_in, const int* in_sizes, int n_in,
                              void* d_out, int out_size, void* d_ws, size_t ws_size,
                              hipStream_t stream) {
    (void)n_in; (void)out_size; (void)d_ws; (void)ws_size;
    const float* x    = (const float*)d_in[0];
    const float* kern = (const float*)d_in[1];
    float* out        = (float*)d_out;

    const int planes = in_sizes[0] / PLANE_IN; // N*C = 8192
    upfirdn2d_down2_kernel<<<planes, 256, 0, stream>>>(x, kern, out);
}